// NeuralstackOnly_54228257079594
// MI455X (gfx1250) — compile-verified
//
#include <hip/hip_runtime.h>
#include <cstdint>

namespace {

constexpr int kB = 32;
constexpr int kT = 256;
constexpr int kV = 2048;
constexpr int kNS = 12;
constexpr int kThreads = 256;          // 8 wave32s
constexpr int kWaves = kThreads / 32;
constexpr float kZeroOffset = 0.001f;
constexpr float kEps = 1e-8f;

__device__ __forceinline__ float eluf(float v) {
  return v > 0.0f ? v : (expf(v) - 1.0f);
}

// Workgroup reduction of NV floats: wave32 shuffle tree, then one LDS slot per
// wave, one barrier, redundant final sum. `slots` must be double-buffered by
// the caller across consecutive uses (no trailing barrier needed then).
template <int NV>
__device__ __forceinline__ void block_reduce(float (&v)[NV], float* slots,
                                             int lane, int wave) {
#pragma unroll
  for (int j = 0; j < NV; ++j) {
    float s = v[j];
#pragma unroll
    for (int off = 16; off > 0; off >>= 1) s += __shfl_xor(s, off, 32);
    if (lane == 0) slots[wave * NV + j] = s;
  }
  __syncthreads();
#pragma unroll
  for (int j = 0; j < NV; ++j) {
    float s = 0.0f;
#pragma unroll
    for (int w = 0; w < kWaves; ++w) s += slots[w * NV + j];
    v[j] = s;
  }
}

}  // namespace

__global__ __launch_bounds__(kThreads)
void neural_stack_scan(const float* __restrict__ x,           // [B,T,V]
                       const float* __restrict__ should_pop,  // [V]
                       const float* __restrict__ sharpen,     // [1]
                       const float* __restrict__ latch_en,    // [V]
                       const float* __restrict__ latch0,      // [B,V]
                       float* __restrict__ outs,              // [B,T,V]
                       float* __restrict__ lats,              // [B,T,V]
                       float* __restrict__ pops,              // [B,T]
                       float* __restrict__ tops) {            // [B,T,V]
  __shared__ float s_x[2][kV];                 // double-buffered input row
  __shared__ float s_red0[kWaves * 4];
  __shared__ float s_redA[2][kWaves * 2];
  __shared__ float s_redB[2][kWaves * 2];

  const int b = blockIdx.x;
  const int tid = threadIdx.x;
  const int lane = tid & 31;
  const int wave = tid >> 5;
  // Each thread owns 8 V-slots: [4*tid .. 4*tid+3] and [1024+4*tid .. +3]
  const int c0 = 4 * tid;
  const int c1 = kV / 2 + 4 * tid;

  float sp[8], le[8], latch[8];
  {
    float4 a0 = *(const float4*)(should_pop + c0);
    float4 a1 = *(const float4*)(should_pop + c1);
    sp[0] = a0.x; sp[1] = a0.y; sp[2] = a0.z; sp[3] = a0.w;
    sp[4] = a1.x; sp[5] = a1.y; sp[6] = a1.z; sp[7] = a1.w;
    float4 b0 = *(const float4*)(latch_en + c0);
    float4 b1 = *(const float4*)(latch_en + c1);
    le[0] = b0.x; le[1] = b0.y; le[2] = b0.z; le[3] = b0.w;
    le[4] = b1.x; le[5] = b1.y; le[6] = b1.z; le[7] = b1.w;
    const float* l0p = latch0 + (size_t)b * kV;
    float4 l0 = *(const float4*)(l0p + c0);
    float4 l1 = *(const float4*)(l0p + c1);
    latch[0] = l0.x; latch[1] = l0.y; latch[2] = l0.z; latch[3] = l0.w;
    latch[4] = l1.x; latch[5] = l1.y; latch[6] = l1.z; latch[7] = l1.w;
  }

  // Stack state lives entirely in VGPRs: 12x8 floats per thread.
  float st[kNS][8];
#pragma unroll
  for (int n = 0; n < kNS; ++n)
#pragma unroll
    for (int k = 0; k < 8; ++k) st[n][k] = kZeroOffset;

  float ptr[kNS];
#pragma unroll
  for (int n = 0; n < kNS; ++n) ptr[n] = (n == 0) ? 1.0f : 0.0f;

  // Pre-loop reduction: {||sp||^2, ||le||^2, dot(latch0,sp), ||latch0||^2}
  float r4[4] = {0.f, 0.f, 0.f, 0.f};
#pragma unroll
  for (int k = 0; k < 8; ++k) {
    r4[0] += sp[k] * sp[k];
    r4[1] += le[k] * le[k];
    r4[2] += latch[k] * sp[k];
    r4[3] += latch[k] * latch[k];
  }
  block_reduce<4>(r4, s_red0, lane, wave);
  const float n_sp = sqrtf(r4[0]);
  const float n_le = sqrtf(r4[1]);
  float dot_l_sp = r4[2];   // dot(latch, should_pop), carried
  float nrm2_l = r4[3];     // ||latch||^2, carried

  const float sharp = sharpen[0];

  // CDNA5 async data mover: each lane copies exactly the 32 bytes it will
  // later read from LDS, so a per-wave s_wait_asynccnt is sufficient.
  auto issue = [&](int t, int buf) {
    const float* g = x + ((size_t)b * kT + t) * kV;
    uint64_t ga0 = (uint64_t)(uintptr_t)(g + c0);
    uint64_t ga1 = (uint64_t)(uintptr_t)(g + c1);
    uint32_t la0 = (uint32_t)(uintptr_t)&s_x[buf][c0];  // low 32b = LDS offset
    uint32_t la1 = (uint32_t)(uintptr_t)&s_x[buf][c1];
    asm volatile("global_load_async_to_lds_b128 %0, %1, off"
                 :: "v"(la0), "v"(ga0) : "memory");
    asm volatile("global_load_async_to_lds_b128 %0, %1, off"
                 :: "v"(la1), "v"(ga1) : "memory");
  };
  issue(0, 0);
  issue(1, 1);

  for (int t = 0; t < kT; ++t) {
    const int buf = t & 1;

    // pop from carried latch statistics (latch at step entry)
    const float cpop =
        dot_l_sp / (fmaxf(n_sp, kEps) * fmaxf(sqrtf(nrm2_l), kEps));
    const float pop = eluf(cpop);
    const float push = 1.0f - pop;

    // Wait for this buffer's async loads (in-order completion: <=2 leaves
    // only the next buffer's two instructions outstanding), then read row.
    if (t + 1 < kT) asm volatile("s_wait_asynccnt 2" ::: "memory");
    else            asm volatile("s_wait_asynccnt 0" ::: "memory");
    float inp[8];
    {
      float4 i0 = *(const float4*)&s_x[buf][c0];
      float4 i1 = *(const float4*)&s_x[buf][c1];
      inp[0] = i0.x; inp[1] = i0.y; inp[2] = i0.z; inp[3] = i0.w;
      inp[4] = i1.x; inp[5] = i1.y; inp[6] = i1.z; inp[7] = i1.w;
    }

    // Phase A reduction: {dot(latch_enable, inp), ||inp||^2}
    float rA[2] = {0.f, 0.f};
#pragma unroll
    for (int k = 0; k < 8; ++k) {
      rA[0] += le[k] * inp[k];
      rA[1] += inp[k] * inp[k];
    }
    block_reduce<2>(rA, s_redA[buf], lane, wave);

    // Re-arm this buffer for t+2. Our LDS reads are retired (values consumed
    // above); explicit dscnt drain guards against async-vs-DS reordering.
    asm volatile("s_wait_dscnt 0" ::: "memory");
    if (t + 2 < kT) issue(t + 2, buf);
    if (t + 4 < kT)  // pull further-ahead row into L2 (global_prefetch_b8)
      __builtin_prefetch(x + ((size_t)b * kT + (t + 4)) * kV + c0, 0, 1);

    const float interp =
        rA[0] / (fmaxf(n_le, kEps) * fmaxf(sqrtf(rA[1]), kEps));

    // Pointer mixture (pre-sharpen) from OLD ptr.
    float mix[kNS];
#pragma unroll
    for (int n = 0; n < kNS; ++n) {
      const float pp = ptr[(n + kNS - 1) % kNS];  // roll +1
      const float pn = ptr[(n + 1) % kNS];        // roll -1
      mix[n] = push * pp + pop * pn;
    }
    // Stack blend using OLD ptr: st = a*st + b*inp + c
#pragma unroll
    for (int n = 0; n < kNS; ++n) {
      const float pp = ptr[(n + kNS - 1) % kNS];
      const float pc = ptr[n];
      const float ca = push * (1.0f - pp) + pop * (1.0f - pc);
      const float cb = push * pp;
      const float cc = pop * kZeroOffset * pc;
#pragma unroll
      for (int k = 0; k < 8; ++k)
        st[n][k] = fmaf(ca, st[n][k], fmaf(cb, inp[k], cc));
    }
    // Sharpen pointer toward one-hot and renormalize.
    float psum = 0.0f;
#pragma unroll
    for (int n = 0; n < kNS; ++n) {
      const float p = powf(fmaxf(mix[n], 0.0f), sharp);
      mix[n] = p;
      psum += p;
    }
    const float pinv = 1.0f / (psum + kEps);
#pragma unroll
    for (int n = 0; n < kNS; ++n) ptr[n] = mix[n] * pinv;

    // top = new_stack . new_ptr (12-tap weighted sum per v)
    float top[8] = {0.f, 0.f, 0.f, 0.f, 0.f, 0.f, 0.f, 0.f};
#pragma unroll
    for (int n = 0; n < kNS; ++n)
#pragma unroll
      for (int k = 0; k < 8; ++k) top[k] = fmaf(st[n][k], ptr[n], top[k]);

    // Latch interpolation.
    const float omi = 1.0f - interp;
#pragma unroll
    for (int k = 0; k < 8; ++k) latch[k] = omi * latch[k] + interp * inp[k];

    // Phase B reduction: next step's {dot(latch,sp), ||latch||^2}.
    float rB[2] = {0.f, 0.f};
#pragma unroll
    for (int k = 0; k < 8; ++k) {
      rB[0] += latch[k] * sp[k];
      rB[1] += latch[k] * latch[k];
    }
    block_reduce<2>(rB, s_redB[buf], lane, wave);
    dot_l_sp = rB[0];
    nrm2_l = rB[1];

    // Coalesced float4 stores of all per-step outputs.
    const size_t base = ((size_t)b * kT + t) * (size_t)kV;
    *(float4*)(outs + base + c0) =
        make_float4(pop * top[0], pop * top[1], pop * top[2], pop * top[3]);
    *(float4*)(outs + base + c1) =
        make_float4(pop * top[4], pop * top[5], pop * top[6], pop * top[7]);
    *(float4*)(lats + base + c0) =
        make_float4(latch[0], latch[1], latch[2], latch[3]);
    *(float4*)(lats + base + c1) =
        make_float4(latch[4], latch[5], latch[6], latch[7]);
    *(float4*)(tops + base + c0) = make_float4(top[0], top[1], top[2], top[3]);
    *(float4*)(tops + base + c1) = make_float4(top[4], top[5], top[6], top[7]);
    if (tid == 0) pops[(size_t)b * kT + t] = pop;
  }
}

extern "C" void kernel_launch(void* const* d_in, const int* in_sizes, int n_in,
                              void* d_out, int out_size, void* d_ws,
                              size_t ws_size, hipStream_t stream) {
  (void)in_sizes; (void)n_in; (void)out_size; (void)d_ws; (void)ws_size;
  const float* x        = (const float*)d_in[0];
  const float* sp       = (const float*)d_in[1];
  const float* sharpen  = (const float*)d_in[2];
  const float* le       = (const float*)d_in[3];
  const float* latch0   = (const float*)d_in[4];

  float* out = (float*)d_out;
  const size_t btv = (size_t)kB * kT * kV;
  float* outs = out;                 // [B,T,V]
  float* lats = out + btv;           // [B,T,V]
  float* pops = out + 2 * btv;       // [B,T]
  float* tops = pops + (size_t)kB * kT;  // [B,T,V]

  hipLaunchKernelGGL(neural_stack_scan, dim3(kB), dim3(kThreads), 0, stream,
                     x, sp, sharpen, le, latch0, outs, lats, pops, tops);
}